// LinearAttention_8211977470154
// MI455X (gfx1250) — compile-verified
//
#include <hip/hip_runtime.h>
#include <math.h>

typedef __attribute__((ext_vector_type(2))) float v2f;
typedef __attribute__((ext_vector_type(4))) float f4;
typedef __attribute__((ext_vector_type(8))) float v8f;

#define N_ 8
#define L_ 8192
#define S_ 8192
#define H_ 8
#define D_ 64
#define EPS_ 1e-6f
#define NCHUNK 16
#define SCHUNK (S_ / NCHUNK)   // 512
#define LSTR 68                 // LDS row stride (floats): 272B, 16B-aligned, conflict-free column reads

// elu(x)+1
__device__ __forceinline__ float fmap(float x) {
    return x > 0.0f ? x + 1.0f : __expf(x);
}

// Async global->LDS copy of 16 bytes per lane (CDNA5 GLOBAL_LOAD_ASYNC_TO_LDS_B128,
// tracked with ASYNCcnt). ldsPtr: generic pointer to __shared__ — low 32 bits of the
// generic encoding are the LDS byte address ({SHARED_BASE, offset} form).
__device__ __forceinline__ void async_copy_b128(void* ldsPtr, const void* gPtr) {
    unsigned lds32 = (unsigned)(unsigned long long)(uintptr_t)ldsPtr;
    unsigned long long g64 = (unsigned long long)(uintptr_t)gPtr;
    asm volatile("global_load_async_to_lds_b128 %0, %1, off"
                 :: "v"(lds32), "v"(g64) : "memory");
}
__device__ __forceinline__ void async_wait0() {
    asm volatile("s_wait_asynccnt 0x0" ::: "memory");
}

__global__ void la_zero_ws(float* __restrict__ ws, int n) {
    int i = blockIdx.x * blockDim.x + threadIdx.x;
    if (i < n) ws[i] = 0.0f;
}

// ---------------------------------------------------------------------------
// Kernel 1: KV[nh][d][v] += sum_s fmap(K[s,d]) * V[s,v], Ksum[nh][d] += sum_s fmap(K[s,d])
// (the reference's /S on V cancels exactly against the *S in the epilogue)
// grid = N*H*NCHUNK blocks, 512 threads (16 waves); wave (dTile,vTile) owns a 16x16 tile.
// ---------------------------------------------------------------------------
__global__ __launch_bounds__(512)
void la_kv_accum(const float* __restrict__ K, const float* __restrict__ V,
                 float* __restrict__ KVws, float* __restrict__ Ksumws)
{
    __shared__ float Ks[32][LSTR];
    __shared__ float Vs[32][LSTR];

    const int bx    = blockIdx.x;
    const int chunk = bx % NCHUNK;
    const int nh    = bx / NCHUNK;
    const int n     = nh / H_;
    const int h     = nh % H_;

    const int tid  = threadIdx.x;
    const int lane = tid & 31;
    const int wid  = tid >> 5;
    const int dTile = wid >> 2;     // 0..3
    const int vTile = wid & 3;      // 0..3
    const int half  = lane >> 4;    // 0 or 1
    const int l16   = lane & 15;

    const size_t rowStride = (size_t)H_ * D_;   // 512 floats
    const float* Kbase = K + (size_t)n * S_ * rowStride + (size_t)h * D_;
    const float* Vbase = V + (size_t)n * S_ * rowStride + (size_t)h * D_;

    // staging: 32 rows x 64 cols, one float4 per thread
    const int sRow = tid >> 4;          // 0..31
    const int sCol = (tid & 15) * 4;    // 0..60

    v8f acc = {};
    float ksum = 0.0f;

    const int sBeg = chunk * SCHUNK;
    const int sEnd = sBeg + SCHUNK;
    for (int s0 = sBeg; s0 < sEnd; s0 += 32) {
        // V: raw async copy global -> LDS (no transform needed)
        async_copy_b128(&Vs[sRow][sCol],
                        Vbase + (size_t)(s0 + sRow) * rowStride + sCol);
        // K: through VGPRs, feature map applied
        {
            const float* kp = Kbase + (size_t)(s0 + sRow) * rowStride + sCol;
            f4 kq = *(const f4*)kp;
            Ks[sRow][sCol + 0] = fmap(kq.x);
            Ks[sRow][sCol + 1] = fmap(kq.y);
            Ks[sRow][sCol + 2] = fmap(kq.z);
            Ks[sRow][sCol + 3] = fmap(kq.w);
        }
        async_wait0();
        __syncthreads();

        if (tid < 64) {
            #pragma unroll
            for (int s = 0; s < 32; ++s) ksum += Ks[s][tid];
        }

        #pragma unroll
        for (int k0 = 0; k0 < 32; k0 += 4) {
            const int ks0 = k0 + 2 * half;
            v2f a, b;
            a.x = Ks[ks0    ][dTile * 16 + l16];   // A(m=d, k=s)
            a.y = Ks[ks0 + 1][dTile * 16 + l16];
            b.x = Vs[ks0    ][vTile * 16 + l16];   // B(k=s, n=v)
            b.y = Vs[ks0 + 1][vTile * 16 + l16];
            acc = __builtin_amdgcn_wmma_f32_16x16x4_f32(
                /*neg_a=*/false, a, /*neg_b=*/false, b,
                /*c_mod=*/(short)0, acc, /*reuse_a=*/false, /*reuse_b=*/false);
        }
        __syncthreads();
    }

    // accumulate partials (16 chunks per (n,h)) into workspace
    float* kvDst = KVws + (size_t)nh * D_ * D_;
    #pragma unroll
    for (int j = 0; j < 8; ++j) {
        const int drow = dTile * 16 + j + 8 * half;   // D-matrix: VGPR j -> M = j + 8*half
        const int vcol = vTile * 16 + l16;
        atomicAdd(&kvDst[drow * D_ + vcol], acc[j]);
    }
    if (tid < 64) atomicAdd(&Ksumws[(size_t)nh * D_ + tid], ksum);
}

// ---------------------------------------------------------------------------
// Kernel 2: out[l,v] = (fmap(Q)[l,:] . KV[:,v]) / (fmap(Q)[l,:] . Ksum + eps)
// grid = N*H*(L/64) blocks, 512 threads (16 waves); wave (mSub,vTile) owns 16x16 tile.
// ---------------------------------------------------------------------------
__global__ __launch_bounds__(512)
void la_out(const float* __restrict__ Q, const float* __restrict__ KVws,
            const float* __restrict__ Ksumws, float* __restrict__ out)
{
    __shared__ float Qs[64][LSTR];
    __shared__ float KVs[64][LSTR];
    __shared__ float KsumS[64];
    __shared__ float Zs[64];

    const int LBLK = L_ / 64;   // 128
    const int bx   = blockIdx.x;
    const int lblk = bx % LBLK;
    const int nh   = bx / LBLK;
    const int n    = nh / H_;
    const int h    = nh % H_;
    const int l0   = lblk * 64;

    const int tid  = threadIdx.x;
    const int lane = tid & 31;
    const int wid  = tid >> 5;
    const int mSub = wid >> 2;
    const int vTile = wid & 3;
    const int half  = lane >> 4;
    const int l16   = lane & 15;

    const size_t rowStride = (size_t)H_ * D_;
    const float* Qbase = Q + ((size_t)n * L_ + l0) * rowStride + (size_t)h * D_;

    // stage KV (contiguous 64x64) with async global->LDS, 2 x b128 per thread
    const float* kvSrc = KVws + (size_t)nh * D_ * D_;
    #pragma unroll
    for (int i = tid; i < 1024; i += 512) {
        const int row = i >> 4;
        const int col = (i & 15) * 4;
        async_copy_b128(&KVs[row][col], kvSrc + row * D_ + col);
    }

    // stage fmap(Q): 64 rows x 64 cols, 2 float4 per thread (through VGPRs)
    const int r0 = tid >> 4;
    const int c0 = (tid & 15) * 4;
    #pragma unroll
    for (int rr = r0; rr < 64; rr += 32) {
        f4 q = *(const f4*)(Qbase + (size_t)rr * rowStride + c0);
        Qs[rr][c0 + 0] = fmap(q.x);
        Qs[rr][c0 + 1] = fmap(q.y);
        Qs[rr][c0 + 2] = fmap(q.z);
        Qs[rr][c0 + 3] = fmap(q.w);
    }
    if (tid < 64) KsumS[tid] = Ksumws[(size_t)nh * D_ + tid];

    async_wait0();
    __syncthreads();

    // per-row normalizer (v_length factor cancelled against V scaling)
    if (tid < 64) {
        float s = 0.0f;
        #pragma unroll
        for (int d = 0; d < 64; ++d) s += Qs[tid][d] * KsumS[d];
        Zs[tid] = 1.0f / (s + EPS_);
    }
    __syncthreads();

    v8f acc = {};
    #pragma unroll
    for (int d0 = 0; d0 < 64; d0 += 4) {
        const int ks0 = d0 + 2 * half;
        v2f a, b;
        a.x = Qs[mSub * 16 + l16][ks0    ];    // A(m=l, k=d) — column read, stride 68 avoids conflicts
        a.y = Qs[mSub * 16 + l16][ks0 + 1];
        b.x = KVs[ks0    ][vTile * 16 + l16];  // B(k=d, n=v)
        b.y = KVs[ks0 + 1][vTile * 16 + l16];
        acc = __builtin_amdgcn_wmma_f32_16x16x4_f32(
            false, a, false, b, (short)0, acc, false, false);
    }

    // epilogue: scale by z and store
    #pragma unroll
    for (int j = 0; j < 8; ++j) {
        const int row = mSub * 16 + j + 8 * half;            // local l
        const int col = vTile * 16 + l16;                    // v
        const float val = acc[j] * Zs[row];
        out[(((size_t)n * L_ + (l0 + row)) * H_ + h) * D_ + col] = val;
    }
}

extern "C" void kernel_launch(void* const* d_in, const int* in_sizes, int n_in,
                              void* d_out, int out_size, void* d_ws, size_t ws_size,
                              hipStream_t stream) {
    (void)in_sizes; (void)n_in; (void)out_size; (void)ws_size;
    const float* Qp = (const float*)d_in[0];
    const float* Kp = (const float*)d_in[1];
    const float* Vp = (const float*)d_in[2];
    float* outp = (float*)d_out;

    // workspace layout (floats): KV [N*H*64*64] then Ksum [N*H*64]
    float* kv   = (float*)d_ws;
    float* ksum = kv + (size_t)N_ * H_ * D_ * D_;
    const int wsFloats = N_ * H_ * D_ * D_ + N_ * H_ * D_;   // 266240 (~1.04 MB)

    la_zero_ws<<<(wsFloats + 255) / 256, 256, 0, stream>>>((float*)d_ws, wsFloats);
    la_kv_accum<<<N_ * H_ * NCHUNK, 512, 0, stream>>>(Kp, Vp, kv, ksum);
    la_out<<<N_ * H_ * (L_ / 64), 512, 0, stream>>>(Qp, kv, ksum, outp);
}